// Bspline_19335942766607
// MI455X (gfx1250) — compile-verified
//
#include <hip/hip_runtime.h>

// CDNA5 / gfx1250: bilinear control-grid upsample expressed as
//   flow_c = (-192 * Wy) * P_c * Wx^T     (per image n, channel c)
// computed with V_WMMA_F32_16X16X4_F32 (full fp32 matrix path, wave32).

typedef __attribute__((ext_vector_type(2))) float v2f;
typedef __attribute__((ext_vector_type(8))) float v8f;

#define GDIM 32     // control grid side
#define HOUT 192    // output H == W
#define TS   16     // WMMA tile side

// Reference interp: q = (i/192)*31; i0 = clip(floor(q),0,30); a = clip(q-i0,0,1)
// Divide replaced by one multiply: exact q = 31*i/192 is never closer than
// 1/192 to an integer (for 0<i<192), while the mul-vs-div rounding delta is
// < 4e-6 -- floor can never flip, alpha shifts by ~1e-6. No IEEE div chain.
__device__ __forceinline__ void interp_coef(int qi, int& i0, float& w0, float& w1) {
    float q = (float)qi * (31.0f / 192.0f);
    float f = floorf(q);
    f = fminf(fmaxf(f, 0.0f), 30.0f);
    i0 = (int)f;
    float a = fminf(fmaxf(q - f, 0.0f), 1.0f);
    w0 = 1.0f - a;
    w1 = a;
}

__global__ __launch_bounds__(128) void bspline_flow_wmma(
    const float* __restrict__ in,   // [N, 32, 32, 2]
    float* __restrict__ out)        // [N, 192, 192, 2]
{
    __shared__ float PY[2 * TS * GDIM];   // per-slab intermediate [c][16][32], 4 KB

    const int yb   = blockIdx.x;          // y-slab 0..11
    const int n    = blockIdx.y;          // image 0..N-1
    const int wave = threadIdx.x >> 5;    // 0..3 (wave32)
    const int lane = threadIdx.x & 31;
    const int half = lane >> 4;           // K-pair selector in A/B fragments
    const int r    = lane & 15;           // M (A) / N (B,C,D) index

    // ---------- Phase A: PY_c = (-192*Wy_slab)(16x32) x P_c(32x32) ----------
    {
        const int c  = wave >> 1;         // channel this wave computes
        const int nt = wave & 1;          // gx half: columns nt*16 .. nt*16+15
        const int y  = yb * TS + r;

        int iy0; float wy0, wy1;
        interp_coef(y, iy0, wy0, wy1);
        wy0 *= -192.0f;                   // fold flow scale (-H == -W == -192)
        wy1 *= -192.0f;

        const float* Pn = in + (size_t)n * (2 * GDIM * GDIM);
        const int j = nt * 16 + r;        // gx column for B fragment

        v8f acc = {};
#pragma unroll
        for (int kk = 0; kk < 8; ++kk) {  // K = 32 in steps of 4
            const int k0 = 4 * kk + 2 * half;
            v2f a, b;
            // A fragment: Wy'[y][k0], Wy'[y][k0+1]  (2 taps per row -> cndmask)
            a.x = (k0     == iy0) ? wy0 : ((k0     == iy0 + 1) ? wy1 : 0.0f);
            a.y = (k0 + 1 == iy0) ? wy0 : ((k0 + 1 == iy0 + 1) ? wy1 : 0.0f);
            // B fragment: P_c[k0][j], P_c[k0+1][j]
            b.x = Pn[((size_t)(k0    ) * GDIM + j) * 2 + c];
            b.y = Pn[((size_t)(k0 + 1) * GDIM + j) * 2 + c];
            acc = __builtin_amdgcn_wmma_f32_16x16x4_f32(
                false, a, false, b, (short)0, acc, false, false);
        }
        // C/D layout: vgpr jr holds rows (jr, jr+8); lanes give N
#pragma unroll
        for (int jr = 0; jr < 8; ++jr) {
            PY[(c * TS + jr + 8 * half) * GDIM + nt * 16 + r] = acc[jr];
        }
    }
    __syncthreads();

    // ---------- Phase B: D_c(tile) = PY_c(16x32) x Wx^T(32x16) ----------
    for (int t = 0; t < 3; ++t) {
        const int xt = wave * 3 + t;      // 12 x-tiles across 4 waves
        const int x  = xt * TS + r;

        int ix0; float wx0, wx1;
        interp_coef(x, ix0, wx0, wx1);

        v8f acc0 = {}, acc1 = {};
#pragma unroll
        for (int kk = 0; kk < 8; ++kk) {
            const int k0 = 4 * kk + 2 * half;
            v2f b;
            b.x = (k0     == ix0) ? wx0 : ((k0     == ix0 + 1) ? wx1 : 0.0f);
            b.y = (k0 + 1 == ix0) ? wx0 : ((k0 + 1 == ix0 + 1) ? wx1 : 0.0f);
            // A fragments from LDS (k0 even -> 8B aligned ds_load_b64)
            v2f a0 = *(const v2f*)&PY[(0 * TS + r) * GDIM + k0];
            v2f a1 = *(const v2f*)&PY[(1 * TS + r) * GDIM + k0];
            acc0 = __builtin_amdgcn_wmma_f32_16x16x4_f32(
                false, a0, false, b, (short)0, acc0, false, false);
            acc1 = __builtin_amdgcn_wmma_f32_16x16x4_f32(
                false, a1, false, b, (short)0, acc1, false, false);
        }

        // Interleave channels per lane -> contiguous 128B per 16-lane half
        float* outx = out + (size_t)n * (HOUT * HOUT * 2)
                          + ((size_t)(yb * TS) * HOUT + x) * 2;
#pragma unroll
        for (int jr = 0; jr < 8; ++jr) {
            const int yloc = jr + 8 * half;
            float2 v;
            v.x = acc0[jr];
            v.y = acc1[jr];
            *(float2*)(outx + (size_t)yloc * HOUT * 2) = v;
        }
    }
}

extern "C" void kernel_launch(void* const* d_in, const int* in_sizes, int n_in,
                              void* d_out, int out_size, void* d_ws, size_t ws_size,
                              hipStream_t stream) {
    const float* in = (const float*)d_in[0];
    float* out = (float*)d_out;
    const int N = in_sizes[0] / (2 * GDIM * GDIM);   // B*T = 400 images
    dim3 grid(HOUT / TS, N);                          // 12 y-slabs x N images
    dim3 block(128);                                  // 4 wave32 waves
    bspline_flow_wmma<<<grid, block, 0, stream>>>(in, out);
}